// CRF_85323820302787
// MI455X (gfx1250) — compile-verified
//
#include <hip/hip_runtime.h>

#define Bn 256
#define Tn 512
#define Kn 128

typedef __attribute__((ext_vector_type(2)))  float    v2f;
typedef __attribute__((ext_vector_type(8)))  float    v8f;

// ---------------------------------------------------------------------------
// Kernel 1: gold-path score: unary + binary, per batch
// ---------------------------------------------------------------------------
__global__ void __launch_bounds__(256) crf_score_kernel(
    const float* __restrict__ logits, const int* __restrict__ tags,
    const int* __restrict__ lengths, const float* __restrict__ trans,
    float* __restrict__ seq_score) {
  __shared__ float sred[256];
  const int b = blockIdx.x;
  const int tid = threadIdx.x;
  const int len = lengths[b];
  const size_t lbase = (size_t)b * Tn * Kn;
  const int tbase = b * Tn;
  float acc = 0.f;
  for (int t = tid; t < len; t += 256) {
    int tg = tags[tbase + t];
    acc += logits[lbase + (size_t)t * Kn + tg];
    if (t >= 1) acc += trans[tags[tbase + t - 1] * Kn + tg];
  }
  sred[tid] = acc;
  __syncthreads();
  for (int s = 128; s > 0; s >>= 1) {
    if (tid < s) sred[tid] += sred[tid + s];
    __syncthreads();
  }
  if (tid == 0) seq_score[b] = sred[0];
}

// ---------------------------------------------------------------------------
// Kernel 2: forward logsumexp scan via f32 WMMA (exp-trick GEMM per step)
// 16 batches / block, 256 threads = 8 waves, wave w owns j-tile [16w,16w+16).
// B-fragments of E = exp(trans) are loop-invariant and live in VGPRs.
// Per step: 1 barrier after (rowmax+exp), 1 barrier after alpha writeback.
// ---------------------------------------------------------------------------
__global__ void __launch_bounds__(256) crf_forward_kernel(
    const float* __restrict__ logits, const int* __restrict__ lengths,
    const float* __restrict__ trans, float* __restrict__ log_norm) {
  __shared__ float alpha_s[16][Kn];
  __shared__ float aexp_s[16][132];   // padded rows: conflict-free frag reads
  __shared__ float mrow_s[16];
  __shared__ int   lens_s[16];
  __shared__ int   maxlen_s;

  const int tid  = threadIdx.x;
  const int lane = tid & 31;
  const int w    = tid >> 5;
  const int b0   = blockIdx.x * 16;
  const int row  = tid >> 4;      // 0..15; the 16 threads of a row share a wave
  const int c0   = tid & 15;

  if (tid < 16) lens_s[tid] = lengths[b0 + tid];

  // alpha0 = logits[:, 0, :]
  #pragma unroll
  for (int k = 0; k < 8; ++k) {
    int j = c0 + 16 * k;
    alpha_s[row][j] = logits[(size_t)(b0 + row) * Tn * Kn + j];
  }
  __syncthreads();
  if (tid == 0) {
    int ml = 0;
    for (int i = 0; i < 16; ++i) ml = max(ml, lens_s[i]);
    maxlen_s = ml;
  }

  const int ncol = w * 16 + (lane & 15);   // output column for this lane
  const int hi   = lane >> 4;              // half-wave id
  const int m    = lane & 15;              // A-matrix row for this lane
  const int mbase = hi * 8;                // C-layout row base
  const int hi2  = hi << 1;

  // B-fragments: v0 = rows {4kk+0 (lanes0-15), 4kk+2 (16-31)}, v1 = rows {+1,+3}
  v2f bf[32];
  #pragma unroll
  for (int kk = 0; kk < 32; ++kk) {
    int kb = 4 * kk + hi2;
    v2f t2;
    t2.x = __expf(trans[kb * Kn + ncol]);
    t2.y = __expf(trans[(kb + 1) * Kn + ncol]);
    bf[kk] = t2;
  }
  __syncthreads();
  const int maxlen = maxlen_s;

  for (int t = 1; t < maxlen; ++t) {
    // prefetch this step's logits for the 8 C rows this lane owns
    float lreg[8];
    #pragma unroll
    for (int r = 0; r < 8; ++r)
      lreg[r] = logits[(size_t)(b0 + mbase + r) * Tn * Kn + (size_t)t * Kn + ncol];

    // phase 1+2: row max via intra-wave butterfly, then aexp = exp(alpha - m)
    float pm = alpha_s[row][c0];
    #pragma unroll
    for (int k = 1; k < 8; ++k) pm = fmaxf(pm, alpha_s[row][c0 + 16 * k]);
    #pragma unroll
    for (int msk = 8; msk >= 1; msk >>= 1)
      pm = fmaxf(pm, __shfl_xor(pm, msk, 16));
    if (c0 == 0) mrow_s[row] = pm;
    #pragma unroll
    for (int k = 0; k < 8; ++k) {
      int j = c0 + 16 * k;
      aexp_s[row][j] = __expf(alpha_s[row][j] - pm);
    }
    __syncthreads();

    // phase 3: (16x128)x(128x16) tile product, two accumulator chains
    v8f acc0 = {}; v8f acc1 = {};
    #pragma unroll
    for (int kk = 0; kk < 32; kk += 2) {
      v2f a0 = *(const v2f*)&aexp_s[m][4 * kk + hi2];
      acc0 = __builtin_amdgcn_wmma_f32_16x16x4_f32(
          false, a0, false, bf[kk], (short)0, acc0, false, false);
      v2f a1 = *(const v2f*)&aexp_s[m][4 * (kk + 1) + hi2];
      acc1 = __builtin_amdgcn_wmma_f32_16x16x4_f32(
          false, a1, false, bf[kk + 1], (short)0, acc1, false, false);
    }

    // phase 4: new alpha = log(sum) + m + logit, masked write
    #pragma unroll
    for (int r = 0; r < 8; ++r) {
      int mm2 = mbase + r;
      if (t < lens_s[mm2]) {
        alpha_s[mm2][ncol] = __logf(acc0[r] + acc1[r]) + mrow_s[mm2] + lreg[r];
      }
    }
    __syncthreads();
  }

  // final log_norm = logsumexp(alpha) per batch, all intra-wave
  float pm = alpha_s[row][c0];
  #pragma unroll
  for (int k = 1; k < 8; ++k) pm = fmaxf(pm, alpha_s[row][c0 + 16 * k]);
  #pragma unroll
  for (int msk = 8; msk >= 1; msk >>= 1)
    pm = fmaxf(pm, __shfl_xor(pm, msk, 16));
  float s = 0.f;
  #pragma unroll
  for (int k = 0; k < 8; ++k) s += __expf(alpha_s[row][c0 + 16 * k] - pm);
  #pragma unroll
  for (int msk = 8; msk >= 1; msk >>= 1)
    s += __shfl_xor(s, msk, 16);
  if (c0 == 0) log_norm[b0 + row] = pm + __logf(s);
}

// ---------------------------------------------------------------------------
// Kernel 3: Viterbi scan (tropical semiring) + in-kernel backtrack
// one batch per block, lane j = destination tag, trans column in registers
// ---------------------------------------------------------------------------
__global__ void __launch_bounds__(128) crf_viterbi_kernel(
    const float* __restrict__ logits, const int* __restrict__ lengths,
    const float* __restrict__ trans, unsigned char* __restrict__ bp,
    float* __restrict__ path_out, float* __restrict__ score_out) {
  __shared__ float v_sh[Kn];
  const int b = blockIdx.x;
  const int j = threadIdx.x;
  const int len = lengths[b];
  const size_t lbase  = (size_t)b * Tn * Kn;
  const size_t bpbase = (size_t)b * Tn * Kn;

  // trans column j, stationary in 128 VGPRs
  float tc[Kn];
  #pragma unroll
  for (int i = 0; i < Kn; ++i) tc[i] = trans[i * Kn + j];

  v_sh[j] = logits[lbase + j];
  __syncthreads();

  float lnext = (len > 1) ? logits[lbase + Kn + j] : 0.f;
  for (int t = 1; t < len; ++t) {
    float lcur = lnext;
    if (t + 1 < len) lnext = logits[lbase + (size_t)(t + 1) * Kn + j];

    float best = v_sh[0] + tc[0];
    int bi = 0;
    #pragma unroll
    for (int i = 1; i < Kn; ++i) {
      float cand = v_sh[i] + tc[i];
      if (cand > best) { best = cand; bi = i; }   // strict >: first-index ties
    }
    bp[bpbase + (size_t)t * Kn + j] = (unsigned char)bi;
    float nv = best + lcur;
    __syncthreads();
    v_sh[j] = nv;
    __syncthreads();
  }

  __syncthreads();
  if (j == 0) {
    float best = v_sh[0]; int bt = 0;
    for (int i = 1; i < Kn; ++i)
      if (v_sh[i] > best) { best = v_sh[i]; bt = i; }
    score_out[b] = best;
    int tag = bt;
    for (int t = Tn - 1; t >= 1; --t) {
      path_out[b * Tn + t] = (float)tag;
      if (t < len) tag = bp[bpbase + (size_t)t * Kn + tag];
    }
    path_out[b * Tn] = (float)tag;
  }
}

// ---------------------------------------------------------------------------
// Kernel 4: loss = sum_b (log_norm - seq_score)
// ---------------------------------------------------------------------------
__global__ void __launch_bounds__(256) crf_loss_kernel(
    const float* __restrict__ seq_score, const float* __restrict__ log_norm,
    float* __restrict__ loss) {
  __shared__ float sred[256];
  const int tid = threadIdx.x;
  sred[tid] = log_norm[tid] - seq_score[tid];
  __syncthreads();
  for (int s = 128; s > 0; s >>= 1) {
    if (tid < s) sred[tid] += sred[tid + s];
    __syncthreads();
  }
  if (tid == 0) loss[0] = sred[0];
}

// ---------------------------------------------------------------------------
extern "C" void kernel_launch(void* const* d_in, const int* in_sizes, int n_in,
                              void* d_out, int out_size, void* d_ws, size_t ws_size,
                              hipStream_t stream) {
  (void)in_sizes; (void)n_in; (void)out_size; (void)ws_size;
  const float* logits  = (const float*)d_in[0];
  const int*   tags    = (const int*)d_in[1];
  const int*   lengths = (const int*)d_in[2];
  const float* trans   = (const float*)d_in[3];

  float* out       = (float*)d_out;
  float* loss_out  = out;                 // [0]
  float* path_out  = out + 1;             // [1 .. 1+B*T)
  float* score_out = out + 1 + Bn * Tn;   // [1+B*T .. +B)

  float* seq_score = (float*)d_ws;                          // 256 f32
  float* log_norm  = seq_score + 256;                       // 256 f32
  unsigned char* bp = (unsigned char*)d_ws + 2048;          // B*T*K bytes

  crf_score_kernel<<<Bn, 256, 0, stream>>>(logits, tags, lengths, trans, seq_score);
  crf_forward_kernel<<<Bn / 16, 256, 0, stream>>>(logits, lengths, trans, log_norm);
  crf_viterbi_kernel<<<Bn, 128, 0, stream>>>(logits, lengths, trans, bp, path_out, score_out);
  crf_loss_kernel<<<1, 256, 0, stream>>>(seq_score, log_norm, loss_out);
}